// SoftVoxelOccupancyVFE_55808805044588
// MI455X (gfx1250) — compile-verified
//
#include <hip/hip_runtime.h>
#include <hip/hip_bf16.h>

typedef float v2f __attribute__((ext_vector_type(2)));
typedef float v8f __attribute__((ext_vector_type(8)));

#define MAX_PTS_NORM 10.0f

// ---- fast reciprocal (v_rcp_f32) ------------------------------------------
#if __has_builtin(__builtin_amdgcn_rcpf)
__device__ __forceinline__ float fast_rcp(float x) { return __builtin_amdgcn_rcpf(x); }
#else
__device__ __forceinline__ float fast_rcp(float x) { return 1.0f / x; }
#endif

// ---- VALU cross-lane butterflies via v_permlane16_b32 ---------------------
#if __has_builtin(__builtin_amdgcn_permlane16_b32)
#define PL16(x, lo, hi) __builtin_amdgcn_permlane16_b32((x), (x), (lo), (hi), false, false)
#define HAVE_PL16 1
#elif __has_builtin(__builtin_amdgcn_permlane16)
#define PL16(x, lo, hi) __builtin_amdgcn_permlane16((x), (x), (lo), (hi), false, false)
#define HAVE_PL16 1
#else
#define HAVE_PL16 0
#endif

// xor-1 add (stays within each 16-lane row)
__device__ __forceinline__ float pair_add(float v) {
#if HAVE_PL16
  return v + __int_as_float(PL16(__float_as_int(v), 0x67452301u, 0xEFCDAB89u));
#else
  return v + __shfl_xor(v, 1, 32);
#endif
}

// full xor butterfly sum within each 16-lane row (result broadcast)
__device__ __forceinline__ float xor_add16(float v) {
#if HAVE_PL16
  v += __int_as_float(PL16(__float_as_int(v), 0x67452301u, 0xEFCDAB89u)); // ^1
  v += __int_as_float(PL16(__float_as_int(v), 0x54761032u, 0xDCFE98BAu)); // ^2
  v += __int_as_float(PL16(__float_as_int(v), 0x32107654u, 0xBA98FEDCu)); // ^4
  v += __int_as_float(PL16(__float_as_int(v), 0xFEDCBA98u, 0x76543210u)); // ^8
  return v;
#else
#pragma unroll
  for (int m = 1; m < 16; m <<= 1) v += __shfl_xor(v, m, 32);
  return v;
#endif
}

__global__ __launch_bounds__(256) void SoftVoxelOccupancyVFE_kernel(
    const float* __restrict__ features,   // [N][32][4]
    const int*   __restrict__ num_points, // [N]
    const float* __restrict__ W1,         // [5][16] row-major
    const float* __restrict__ b1,         // [16]
    const float* __restrict__ W2,         // [16]
    const float* __restrict__ b2,         // [1]
    float* __restrict__ out,              // [N]
    int N)
{
  const int lane          = threadIdx.x & 31;
  const int waveInBlock   = threadIdx.x >> 5;
  const int wavesPerBlock = blockDim.x >> 5;
  const int tile  = blockIdx.x * wavesPerBlock + waveInBlock; // 16 voxels / wave
  const int vbase = tile * 16;
  if (vbase >= N) return;   // wave-uniform: EXEC stays all-ones below

  // ---- lane roles: 2 lanes per voxel, 16 points each ---------------------
  const int  iVox  = lane >> 1;        // voxel index within tile (0..15)
  const int  half  = lane & 1;         // which 16-point half this lane sums
  const int  v     = vbase + iVox;
  const int  vc    = (v < N) ? v : (N - 1);
  const int  np    = (v < N) ? num_points[vc] : 0;
  const float npf  = (float)np;
  const int  pbase = half << 4;

  // ---- one-pass masked sufficient statistics -----------------------------
  // wave streams a contiguous 8KB tile: 16 independent b128 loads per lane
  const float4* __restrict__ src =
      (const float4*)features + ((long long)vc * 32 + pbase);

  float sx = 0.0f, sy = 0.0f, sz = 0.0f, ss = 0.0f;
#pragma unroll
  for (int p = 0; p < 16; ++p) {
    const float4 f = src[p];
    const float  m = (pbase + p < np) ? 1.0f : 0.0f;
    const float  x = f.x * m, y = f.y * m, z = f.z * m;
    sx += x; sy += y; sz += z;
    ss = fmaf(x, x, fmaf(y, y, fmaf(z, z, ss)));
  }
  // combine the two half-lanes of each voxel (xor-1, VALU permlane)
  sx = pair_add(sx); sy = pair_add(sy); sz = pair_add(sz); ss = pair_add(ss);

  // ---- per-voxel features -------------------------------------------------
  const float denom = fmaxf(npf, 1.0f);
  const float rden  = fast_rcp(denom);
  const float mx = sx * rden, my = sy * rden, mz = sz * rden;
  // sum_c var_c = ss - 2*(m.s) + |m|^2 * n   (one-pass form of reference)
  const float q = ss - 2.0f * (mx * sx + my * sy + mz * sz)
                + (mx * mx + my * my + mz * mz) * npf;
  const float var_mean = fmaxf(q, 0.0f) * rden * (1.0f / 3.0f);
  const float p_var    = __expf(-0.5f * var_mean);
  const float p_den    = fminf(npf, MAX_PTS_NORM) * (1.0f / MAX_PTS_NORM);

  // ---- gather into WMMA A fragment layout (3 ds_bpermute total) ----------
  // A 16x16x4 layout: lane t<16 holds row t K0/K1 (af0) and K4/K5 (af1);
  //                   lane t>=16 holds row t-16 K2/K3.
  // voxel i's stats live in lanes 2i (even) and 2i+1 (odd):
  //   even lanes serve targets t=i      -> need {p_den, p_var, mz}
  //   odd  lanes serve targets t=i+16   -> need {mx, my, 0}
  const float val0 = half ? mx   : p_den;
  const float val1 = half ? my   : p_var;
  const float val2 = half ? 0.0f : mz;
  const int   srcLane = (lane < 16) ? (2 * lane) : (2 * lane - 31);
  v2f af0, af1;
  af0[0] = __shfl(val0, srcLane, 32);
  af0[1] = __shfl(val1, srcLane, 32);
  af1[0] = __shfl(val2, srcLane, 32);
  af1[1] = 0.0f;

  // ---- B fragments for W1 (5x16, K padded to 8) --------------------------
  const int  n      = lane & 15;
  const bool hiHalf = lane >= 16;
  v2f bf0, bf1;
  bf0[0] = hiHalf ? W1[2 * 16 + n] : W1[0 * 16 + n];
  bf0[1] = hiHalf ? W1[3 * 16 + n] : W1[1 * 16 + n];
  bf1[0] = hiHalf ? 0.0f           : W1[4 * 16 + n];
  bf1[1] = 0.0f;

  const float w2n   = W2[n];
  const float bias2 = b2[0];

  // C preloaded with layer-1 bias (broadcast per column)
  v8f acc;
#pragma unroll
  for (int r = 0; r < 8; ++r) acc[r] = b1[n];

  // ---- layer 1 on the matrix core: D = A(16x8) x W1(8x16) + b1 -----------
  acc = __builtin_amdgcn_wmma_f32_16x16x4_f32(
      false, af0, false, bf0, (short)0, acc, false, false);
  acc = __builtin_amdgcn_wmma_f32_16x16x4_f32(
      false, af1, false, bf1, (short)0, acc, false, false);

  // ---- relu + layer 2 (16 -> 1) + sigmoid + store ------------------------
  // D layout: component r, lanes 0-15 = row r, lanes 16-31 = row r+8
#pragma unroll
  for (int r = 0; r < 8; ++r) {
    const float h = fmaxf(acc[r], 0.0f);
    const float s = xor_add16(h * w2n) + bias2;          // VALU butterfly
    const float sig = fast_rcp(1.0f + __expf(-s));
    if (n == 0) {              // lane 0 stores rows 0-7, lane 16 rows 8-15
      const int row = hiHalf ? (8 + r) : r;
      const int vv  = vbase + row;
      if (vv < N) out[vv] = sig;
    }
  }
}

extern "C" void kernel_launch(void* const* d_in, const int* in_sizes, int n_in,
                              void* d_out, int out_size, void* d_ws, size_t ws_size,
                              hipStream_t stream) {
  // setup_inputs order: features, num_points, coors, W1, b1, W2, b2
  const float* features   = (const float*)d_in[0];
  const int*   num_points = (const int*)  d_in[1];
  // d_in[2] = coors : unused by the reference computation
  const float* W1 = (const float*)d_in[3];
  const float* b1 = (const float*)d_in[4];
  const float* W2 = (const float*)d_in[5];
  const float* b2 = (const float*)d_in[6];
  float* out = (float*)d_out;

  const int N = in_sizes[1];           // num_points has N elements
  const int voxelsPerBlock = 128;      // 8 waves x 16 voxels
  const int blocks = (N + voxelsPerBlock - 1) / voxelsPerBlock;
  SoftVoxelOccupancyVFE_kernel<<<blocks, 256, 0, stream>>>(
      features, num_points, W1, b1, W2, b2, out, N);
}